// AModel_69277822484840
// MI455X (gfx1250) — compile-verified
//
#include <hip/hip_runtime.h>

typedef __attribute__((ext_vector_type(2))) float v2f;
typedef __attribute__((ext_vector_type(8))) float v8f;

#define kB 16
#define kN 170
#define kF 64
#define kT 150
#define kO 64

static __device__ inline v8f wmma4(v2f a, v2f b, v8f c) {
  // V_WMMA_F32_16X16X4_F32 : D = A(16x4) * B(4x16) + C
  return __builtin_amdgcn_wmma_f32_16x16x4_f32(
      /*neg_a=*/false, a, /*neg_b=*/false, b,
      /*c_mod=*/(short)0, c, /*reuse_a=*/false, /*reuse_b=*/false);
}

enum { EPI_STORE = 0, EPI_BIAS_SIGMOID = 1, EPI_ACC = 2, EPI_ACC_RELU = 3 };

// ---------------------------------------------------------------------------
// Generic single-wave fp32 WMMA GEMM: C[M x Nn] = A[M x K] * B[K x Nn]
// A element (m,k) at A[m*sAm + k*sAk]; B element (k,n) at B[k*sBk + n*sBn].
// Branch-free edge handling: loads always hit a clamped in-bounds address,
// out-of-range contributions are zeroed with a 0/1 mask multiply.
// ---------------------------------------------------------------------------
template <int EPI>
__global__ __launch_bounds__(32) void wmma_gemm(
    const float* __restrict__ A, long sAm, long sAk, long sAbat,
    const float* __restrict__ Bm, long sBk, long sBn, long sBbat,
    float* __restrict__ C, long ldc, long sCbat,
    const float* __restrict__ bias, long ldbias,
    int M, int Nn, int K) {
  const int bat = blockIdx.z;
  A += (long)bat * sAbat;
  Bm += (long)bat * sBbat;
  C += (long)bat * sCbat;
  const int lane = threadIdx.x;
  const int half = lane >> 4;   // 0: K 0,1  | 1: K 2,3
  const int l = lane & 15;
  const int m0 = blockIdx.y * 16;
  const int n0 = blockIdx.x * 16;
  const int am = m0 + l;  // A row held by this lane
  const int bn = n0 + l;  // B col held by this lane

  // clamped (always in-bounds) row/col + zero masks for the padding region
  const int amc = (am < M) ? am : (M - 1);
  const int bnc = (bn < Nn) ? bn : (Nn - 1);
  const float mA = (am < M) ? 1.f : 0.f;
  const float mB = (bn < Nn) ? 1.f : 0.f;

  const float* pA = A + (long)amc * sAm + (long)(half * 2) * sAk;
  const float* pB = Bm + (long)bnc * sBn + (long)(half * 2) * sBk;
  const long dA = 4 * sAk, dB = 4 * sBk;

  v8f acc = {0.f, 0.f, 0.f, 0.f, 0.f, 0.f, 0.f, 0.f};
  const int Kfull = K >> 2;  // guard-free groups of 4
  for (int kk = 0; kk < Kfull; ++kk) {
    v2f a, b;
    a.x = pA[0] * mA;
    a.y = pA[sAk] * mA;
    b.x = pB[0] * mB;
    b.y = pB[sBk] * mB;
    pA += dA;
    pB += dB;
    acc = wmma4(a, b, acc);
  }
  if (K & 3) {  // single clamped tail group (k = 4*Kfull .. 4*Kfull+3)
    const int k0 = Kfull * 4 + half * 2;
    const int k0c = (k0 < K) ? k0 : (K - 1);
    const int k1c = (k0 + 1 < K) ? (k0 + 1) : (K - 1);
    const float av0 = A[(long)amc * sAm + (long)k0c * sAk];
    const float av1 = A[(long)amc * sAm + (long)k1c * sAk];
    const float bv0 = Bm[(long)bnc * sBn + (long)k0c * sBk];
    const float bv1 = Bm[(long)bnc * sBn + (long)k1c * sBk];
    v2f a, b;
    a.x = (k0 < K) ? av0 * mA : 0.f;
    a.y = (k0 + 1 < K) ? av1 * mA : 0.f;
    b.x = (k0 < K) ? bv0 * mB : 0.f;
    b.y = (k0 + 1 < K) ? bv1 * mB : 0.f;
    acc = wmma4(a, b, acc);
  }

#pragma unroll
  for (int v = 0; v < 8; ++v) {
    const int m = m0 + v + half * 8;
    const int n = n0 + l;
    if (m < M && n < Nn) {
      float z = acc[v];
      const long ci = (long)m * ldc + n;
      if constexpr (EPI == EPI_BIAS_SIGMOID) {
        z += bias[(long)m * ldbias + n];
        z = 1.0f / (1.0f + expf(-z));
      } else if constexpr (EPI == EPI_ACC) {
        z += C[ci];
      } else if constexpr (EPI == EPI_ACC_RELU) {
        z += C[ci];
        z = fmaxf(z, 0.f);
      }
      C[ci] = z;
    }
  }
}

// ---------------------------------------------------------------------------
// Pass over x: rhs_t[b,n,t] = sum_f U3[f]*x ; xw3[b,n,t] = sum_f W3[f]*x
// ---------------------------------------------------------------------------
__global__ void reduce_f_kernel(const float* __restrict__ x,
                                const float* __restrict__ U3,
                                const float* __restrict__ W3,
                                float* __restrict__ rhs_t,
                                float* __restrict__ xw3) {
  const int idx = blockIdx.x * blockDim.x + threadIdx.x;  // (b*N+n)*T + t
  if (idx >= kB * kN * kT) return;
  const int t = idx % kT;
  const int bn = idx / kT;
  const float* xp = x + (long)bn * kF * kT + t;
  float s3 = 0.f, sw = 0.f;
  for (int f = 0; f < kF; ++f) {
    const float v = xp[(long)f * kT];
    s3 += U3[f] * v;
    sw += W3[f] * v;
  }
  rhs_t[idx] = s3;
  xw3[idx] = sw;
}

// xu1[b,t,f] = sum_n U1[n] * x[b,n,f,t]
__global__ void reduce_n_kernel(const float* __restrict__ x,
                                const float* __restrict__ U1,
                                float* __restrict__ xu1) {
  const int idx = blockIdx.x * blockDim.x + threadIdx.x;  // (b*T+t)*F + f
  if (idx >= kB * kT * kF) return;
  const int f = idx % kF;
  const int bt = idx / kF;
  const int t = bt % kT;
  const int b = bt / kT;
  float s = 0.f;
  for (int n = 0; n < kN; ++n)
    s += U1[n] * x[(((long)(b * kN + n)) * kF + f) * kT + t];
  xu1[idx] = s;
}

// EW1[b,t] = sum_u E[b,t,u] * W1[u]
__global__ void ew1_kernel(const float* __restrict__ E,
                           const float* __restrict__ W1,
                           float* __restrict__ EW1) {
  const int idx = blockIdx.x * blockDim.x + threadIdx.x;  // b*T + t
  if (idx >= kB * kT) return;
  const float* p = E + (long)idx * kT;
  float s = 0.f;
  for (int u = 0; u < kT; ++u) s += p[u] * W1[u];
  EW1[idx] = s;
}

// zs[b,n,f] = sum_t x[b,n,f,t] * EW1[b,t]
__global__ void zs_kernel(const float* __restrict__ x,
                          const float* __restrict__ EW1,
                          float* __restrict__ zs) {
  const int idx = blockIdx.x * blockDim.x + threadIdx.x;  // (b*N+n)*F + f
  if (idx >= kB * kN * kF) return;
  const int b = idx / (kN * kF);
  const float* xp = x + (long)idx * kT;
  const float* ep = EW1 + (long)b * kT;
  float s = 0.f;
  for (int t = 0; t < kT; ++t) s += xp[t] * ep[t];
  zs[idx] = s;
}

// softmax over middle axis: data (BATCH, L, C), softmax over L for fixed (bat,c)
__global__ void softmax_mid_kernel(float* __restrict__ data, int L, int C,
                                   int total) {
  const int idx = blockIdx.x * blockDim.x + threadIdx.x;  // bat*C + c
  if (idx >= total) return;
  const int c = idx % C;
  const int bat = idx / C;
  float* p = data + (long)bat * L * C + c;
  float mx = -3.4e38f;
  for (int i = 0; i < L; ++i) mx = fmaxf(mx, p[(long)i * C]);
  float s = 0.f;
  for (int i = 0; i < L; ++i) {
    const float e = expf(p[(long)i * C] - mx);
    p[(long)i * C] = e;
    s += e;
  }
  const float inv = 1.f / s;
  for (int i = 0; i < L; ++i) p[(long)i * C] *= inv;
}

// TkS[kc,b,n,m] = Tk[kc,n,m] * S[b,n,m]
__global__ void tks_kernel(const float* __restrict__ Tk,
                           const float* __restrict__ S,
                           float* __restrict__ TkS) {
  const int idx = blockIdx.x * blockDim.x + threadIdx.x;
  if (idx >= 3 * kB * kN * kN) return;
  const int m = idx % kN;
  const int n = (idx / kN) % kN;
  const int b = (idx / (kN * kN)) % kB;
  const int kc = idx / (kN * kN * kB);
  TkS[idx] = Tk[((long)kc * kN + n) * kN + m] * S[((long)b * kN + n) * kN + m];
}

// ---------------------------------------------------------------------------
// Final fused: time conv (1x3) + residual 1x1 + ReLU + LayerNorm(channel) ->
// out (B,N,O,T). One workgroup = (b, n, t-tile16); 4 waves, wave w -> o-tile w.
// Conv decomposed into 3 shifted GEMMs + residual GEMM, all via fp32 WMMA.
// Loads are branch-free (clamped address + mask multiply).
// ---------------------------------------------------------------------------
__global__ __launch_bounds__(128) void final_kernel(
    const float* __restrict__ x, const float* __restrict__ sg,
    const float* __restrict__ tc_w, const float* __restrict__ tc_b,
    const float* __restrict__ rc_w, const float* __restrict__ rc_b,
    const float* __restrict__ ln_w, const float* __restrict__ ln_b,
    float* __restrict__ out) {
  const int tt = blockIdx.x;  // t tile (10)
  const int n = blockIdx.y;   // node (170)
  const int b = blockIdx.z;   // batch (16)
  const int wave = threadIdx.x >> 5;
  const int lane = threadIdx.x & 31;
  const int half = lane >> 4;
  const int l = lane & 15;
  const int o0 = wave * 16;
  const int t0 = tt * 16;
  const int am = o0 + l;  // output channel row held by this lane
  const int bt = t0 + l;  // t column held by this lane

  const float* xb = x + ((long)(b * kN + n) * kF) * kT;    // (f,t) ld=T
  const float* sgb = sg + ((long)(b * kN + n) * kO) * kT;  // (i,t) ld=T

  v8f acc = {0.f, 0.f, 0.f, 0.f, 0.f, 0.f, 0.f, 0.f};

  // residual 1x1: rc_w (O x F) @ x[b,n] (F x T)
  const int btc = (bt < kT) ? bt : (kT - 1);
  const float mT = (bt < kT) ? 1.f : 0.f;
  {
    const float* pA = rc_w + am * kF + half * 2;
    const float* pB = xb + (long)(half * 2) * kT + btc;
    for (int kk = 0; kk < kF / 4; ++kk) {
      v2f a, bb;
      a.x = pA[0];
      a.y = pA[1];
      bb.x = pB[0] * mT;
      bb.y = pB[kT] * mT;
      pA += 4;
      pB += 4 * kT;
      acc = wmma4(a, bb, acc);
    }
  }
  // time conv: sum_d  tc_w[:,:,0,d] (O x I) @ sg[b,n,:, t+d-1] (I x T)
  for (int d = 0; d < 3; ++d) {
    const int ts = bt + d - 1;
    const int tsc = (ts < 0) ? 0 : ((ts < kT) ? ts : (kT - 1));
    const float mS = (ts >= 0 && ts < kT) ? 1.f : 0.f;
    const float* pA = tc_w + (am * kO + half * 2) * 3 + d;
    const float* pB = sgb + (long)(half * 2) * kT + tsc;
    for (int kk = 0; kk < kO / 4; ++kk) {
      v2f a, bb;
      a.x = pA[0];
      a.y = pA[3];
      bb.x = pB[0] * mS;
      bb.y = pB[kT] * mS;
      pA += 12;
      pB += 4 * kT;
      acc = wmma4(a, bb, acc);
    }
  }

  // bias + ReLU, then LayerNorm over the 64 channels for each t column
  float yv[8];
  float s1 = 0.f, s2 = 0.f;
#pragma unroll
  for (int v = 0; v < 8; ++v) {
    const int o = o0 + v + half * 8;
    float y = acc[v] + tc_b[o] + rc_b[o];
    y = fmaxf(y, 0.f);
    yv[v] = y;
    s1 += y;
    s2 += y * y;
  }
  // lanes l and l+16 hold the same t column -> pair reduce (16 channels/wave)
  s1 += __shfl_xor(s1, 16, 32);
  s2 += __shfl_xor(s2, 16, 32);
  __shared__ float red1[4][16];
  __shared__ float red2[4][16];
  if (half == 0) {
    red1[wave][l] = s1;
    red2[wave][l] = s2;
  }
  __syncthreads();
  const float tot1 = red1[0][l] + red1[1][l] + red1[2][l] + red1[3][l];
  const float tot2 = red2[0][l] + red2[1][l] + red2[2][l] + red2[3][l];
  const float mean = tot1 * (1.f / 64.f);
  const float var = tot2 * (1.f / 64.f) - mean * mean;
  const float rs = rsqrtf(var + 1e-5f);
  if (bt < kT) {
#pragma unroll
    for (int v = 0; v < 8; ++v) {
      const int o = o0 + v + half * 8;
      const float y = (yv[v] - mean) * rs * ln_w[o] + ln_b[o];
      out[((long)(b * kN + n) * kO + o) * kT + bt] = y;
    }
  }
}

// ---------------------------------------------------------------------------
extern "C" void kernel_launch(void* const* d_in, const int* in_sizes, int n_in,
                              void* d_out, int out_size, void* d_ws,
                              size_t ws_size, hipStream_t stream) {
  (void)in_sizes; (void)n_in; (void)out_size; (void)ws_size;
  const float* x = (const float*)d_in[0];
  const float* U1 = (const float*)d_in[1];
  const float* U2 = (const float*)d_in[2];
  const float* U3 = (const float*)d_in[3];
  const float* be = (const float*)d_in[4];
  const float* Ve = (const float*)d_in[5];
  const float* W1 = (const float*)d_in[6];
  const float* W2 = (const float*)d_in[7];
  const float* W3 = (const float*)d_in[8];
  const float* bs = (const float*)d_in[9];
  const float* Vs = (const float*)d_in[10];
  const float* Tk = (const float*)d_in[11];
  const float* Theta = (const float*)d_in[12];
  const float* tc_w = (const float*)d_in[13];
  const float* tc_b = (const float*)d_in[14];
  const float* rc_w = (const float*)d_in[15];
  const float* rc_b = (const float*)d_in[16];
  const float* ln_w = (const float*)d_in[17];
  const float* ln_b = (const float*)d_in[18];
  float* out = (float*)d_out;

  // workspace carve-up (floats)
  float* ws = (float*)d_ws;
  size_t off = 0;
  auto alloc = [&](size_t n) { float* p = ws + off; off += n; return p; };
  float* rhs_t = alloc((size_t)kB * kN * kT);       // (B,N,T)
  float* xw3   = alloc((size_t)kB * kN * kT);       // (B,N,T)
  float* xu1   = alloc((size_t)kB * kT * kF);       // (B,T,F)
  float* lhs_t = alloc((size_t)kB * kT * kN);       // (B,T,N)
  float* Pbuf  = alloc((size_t)kB * kT * kT);       // sigmoid(E_pre)
  float* Ebuf  = alloc((size_t)kB * kT * kT);       // Ve@P -> softmax -> E
  float* EW1   = alloc((size_t)kB * kT);
  float* zsb   = alloc((size_t)kB * kN * kF);       // (B,N,F)
  float* lhs_s = alloc((size_t)kB * kN * kT);       // (B,N,T)
  float* rhs_s = alloc((size_t)kB * kT * kN);       // (B,T,N)
  float* Psig  = alloc((size_t)kB * kN * kN);
  float* Sbuf  = alloc((size_t)kB * kN * kN);       // Vs@Psig -> softmax -> S
  float* TkS   = alloc((size_t)3 * kB * kN * kN);
  float* aggb  = alloc((size_t)kB * kN * kF * kT);  // (B,N,F,T)
  float* outb  = alloc((size_t)kB * kN * kO * kT);  // spatial_gcn (B,N,O,T)

  const int THR = 256;
  // 1. fused f-reductions over x
  reduce_f_kernel<<<(kB * kN * kT + THR - 1) / THR, THR, 0, stream>>>(
      x, U3, W3, rhs_t, xw3);
  // 2. n-reduction over x
  reduce_n_kernel<<<(kB * kT * kF + THR - 1) / THR, THR, 0, stream>>>(
      x, U1, xu1);
  // 3. lhs_t[b] (T x N) = xu1[b] (T x F) @ U2 (F x N)
  wmma_gemm<EPI_STORE><<<dim3(11, 10, kB), 32, 0, stream>>>(
      xu1, kF, 1, (long)kT * kF, U2, kN, 1, 0,
      lhs_t, kN, (long)kT * kN, nullptr, 0, kT, kN, kF);
  // 4. P[b] = sigmoid(lhs_t[b] @ rhs_t[b] + be)   (T x T, K=N)
  wmma_gemm<EPI_BIAS_SIGMOID><<<dim3(10, 10, kB), 32, 0, stream>>>(
      lhs_t, kN, 1, (long)kT * kN, rhs_t, kT, 1, (long)kN * kT,
      Pbuf, kT, (long)kT * kT, be, kT, kT, kT, kN);
  // 5. E2[b] = Ve @ P[b]   (T x T, K=T)
  wmma_gemm<EPI_STORE><<<dim3(10, 10, kB), 32, 0, stream>>>(
      Ve, kT, 1, 0, Pbuf, kT, 1, (long)kT * kT,
      Ebuf, kT, (long)kT * kT, nullptr, 0, kT, kT, kT);
  // 6. softmax over axis=1 (rows, per column)
  softmax_mid_kernel<<<(kB * kT + THR - 1) / THR, THR, 0, stream>>>(
      Ebuf, kT, kT, kB * kT);
  // 7. EW1
  ew1_kernel<<<(kB * kT + THR - 1) / THR, THR, 0, stream>>>(Ebuf, W1, EW1);
  // 8. zs
  zs_kernel<<<(kB * kN * kF + THR - 1) / THR, THR, 0, stream>>>(x, EW1, zsb);
  // 9. lhs_s[b] (N x T) = zs[b] (N x F) @ W2 (F x T)
  wmma_gemm<EPI_STORE><<<dim3(10, 11, kB), 32, 0, stream>>>(
      zsb, kF, 1, (long)kN * kF, W2, kT, 1, 0,
      lhs_s, kT, (long)kN * kT, nullptr, 0, kN, kT, kF);
  // 10. rhs_s[b] (T x N): rhs_s[b,u,n] = sum_t E[b,t,u] * xw3[b,n,t]
  wmma_gemm<EPI_STORE><<<dim3(11, 10, kB), 32, 0, stream>>>(
      Ebuf, 1, kT, (long)kT * kT,        // A(m=u,k=t) = E[b,t,u]
      xw3, 1, kT, (long)kN * kT,         // B(k=t,n)   = xw3[b,n,t]
      rhs_s, kN, (long)kT * kN, nullptr, 0, kT, kN, kT);
  // 11. Psig[b] = sigmoid(lhs_s[b] @ rhs_s[b] + bs)  (N x N, K=T)
  wmma_gemm<EPI_BIAS_SIGMOID><<<dim3(11, 11, kB), 32, 0, stream>>>(
      lhs_s, kT, 1, (long)kN * kT, rhs_s, kN, 1, (long)kT * kN,
      Psig, kN, (long)kN * kN, bs, kN, kN, kN, kT);
  // 12. S2[b] = Vs @ Psig[b]
  wmma_gemm<EPI_STORE><<<dim3(11, 11, kB), 32, 0, stream>>>(
      Vs, kN, 1, 0, Psig, kN, 1, (long)kN * kN,
      Sbuf, kN, (long)kN * kN, nullptr, 0, kN, kN, kN);
  // 13. softmax over axis=1
  softmax_mid_kernel<<<(kB * kN + THR - 1) / THR, THR, 0, stream>>>(
      Sbuf, kN, kN, kB * kN);
  // 14. TkS
  tks_kernel<<<(3 * kB * kN * kN + THR - 1) / THR, THR, 0, stream>>>(
      Tk, Sbuf, TkS);
  // 15. Chebyshev conv: per k, agg = TkS_k^T @ x ; out (+)= Theta_k^T @ agg
  for (int kc = 0; kc < 3; ++kc) {
    // agg[b] (N x F*T) = TkS_k[b]^T (N x N) @ x[b] (N x F*T)
    wmma_gemm<EPI_STORE><<<dim3(600, 11, kB), 32, 0, stream>>>(
        TkS + (size_t)kc * kB * kN * kN, 1, kN, (long)kN * kN,  // A(m,k=n)=TkS[k,b,n,m]
        x, (long)kF * kT, 1, (long)kN * kF * kT,
        aggb, (long)kF * kT, (long)kN * kF * kT, nullptr, 0,
        kN, kF * kT, kN);
    // out[b,m] (O x T) (+)= Theta_k^T (O x F) @ agg[b,m] (F x T)
    const dim3 g9(10, 4, kB * kN);
    const float* Th = Theta + (size_t)kc * kF * kO;
    if (kc == 0)
      wmma_gemm<EPI_STORE><<<g9, 32, 0, stream>>>(
          Th, 1, kO, 0, aggb, kT, 1, (long)kF * kT,
          outb, kT, (long)kO * kT, nullptr, 0, kO, kT, kF);
    else if (kc == 1)
      wmma_gemm<EPI_ACC><<<g9, 32, 0, stream>>>(
          Th, 1, kO, 0, aggb, kT, 1, (long)kF * kT,
          outb, kT, (long)kO * kT, nullptr, 0, kO, kT, kF);
    else
      wmma_gemm<EPI_ACC_RELU><<<g9, 32, 0, stream>>>(
          Th, 1, kO, 0, aggb, kT, 1, (long)kF * kT,
          outb, kT, (long)kO * kT, nullptr, 0, kO, kT, kF);
  }
  // 16. fused time-conv + residual + ReLU + LayerNorm
  final_kernel<<<dim3(10, kN, kB), 128, 0, stream>>>(
      x, outb, tc_w, tc_b, rc_w, rc_b, ln_w, ln_b, out);
}